// Pooling_21577915695109
// MI455X (gfx1250) — compile-verified
//
#include <hip/hip_runtime.h>

// x:     (B=4, N=50000, H=64) f32
// index: (N=50000, M=16)      int, values in [0, N]; N means implicit zero row
// out:   (B=4, N=50000, H=64) f32 ; out[b,n,:] = max_m xpad[b, index[n,m], :]
//
// One half-wave (16 lanes) per node, lane j owns float4 chunk j of H=64.
// m-outer / batch-inner: one DPP index broadcast + one 32-bit offset MAD per
// neighbor feeds four saddr-form b128 gathers (one per batch slab).
// Padded indices are remapped once per node to a valid duplicate (idempotent
// under max); their zero contribution is restored via a post-loop floor.

typedef float v4f __attribute__((ext_vector_type(4)));

constexpr int HID   = 64;
constexpr int MN    = 16;
constexpr int BATCH = 4;
constexpr int NODES_PER_BLOCK = 16;   // 256 threads / 16 lanes-per-node

__device__ __forceinline__ v4f vmax4(v4f a, v4f b) {
    v4f r;
    r.x = fmaxf(a.x, b.x);
    r.y = fmaxf(a.y, b.y);
    r.z = fmaxf(a.z, b.z);
    r.w = fmaxf(a.w, b.w);
    return r;
}

__global__ __launch_bounds__(256) void gather_max16_kernel(
    const float* __restrict__ x,
    const int*   __restrict__ index,
    float*       __restrict__ out,
    int N)
{
    const int tid    = threadIdx.x;
    const int lane16 = tid & 15;                        // float4 chunk within H
    const int node   = blockIdx.x * NODES_PER_BLOCK + (tid >> 4);
    if (node >= N) return;

    int myIdx = __builtin_nontemporal_load(index + (size_t)node * MN + lane16);

    // Validity bookkeeping, once per node (per half-wave).
    const bool     valid    = (myIdx < N);
    const unsigned ballot   = (unsigned)__ballot(valid);
    const unsigned halfMask = (ballot >> (tid & 16)) & 0xFFFFu;

    const unsigned laneOff = (unsigned)lane16 * 16u;                 // bytes in row
    const unsigned outOff  = (unsigned)node * (HID * 4u) + laneOff;  // bytes in slab
    const size_t   slab    = (size_t)N * HID * sizeof(float);        // one batch
    const char* xb = (const char*)x;
    char*       ob = (char*)out;

    if (halfMask == 0u) {
        // All 16 neighbors padded: max over 16 zero rows == 0. (~(1/50001)^16)
        v4f z = {0.f, 0.f, 0.f, 0.f};
        #pragma unroll
        for (int b = 0; b < BATCH; ++b)
            __builtin_nontemporal_store(z, (v4f*)(ob + (size_t)b * slab + outOff));
        return;
    }

    // Remap padded indices to the node's first valid neighbor; duplicates do
    // not change the max. Zero contribution of padded rows restored below.
    const int firstValid = __ffs(halfMask) - 1;
    const int rep        = __shfl(myIdx, firstValid, 16);
    myIdx = valid ? myIdx : rep;

    const float floorVal = (halfMask == 0xFFFFu) ? -__builtin_huge_valf() : 0.0f;

    v4f a0, a1, a2, a3;
    a0.x = a0.y = a0.z = a0.w = -__builtin_huge_valf();
    a1 = a0; a2 = a0; a3 = a0;

    #pragma unroll
    for (int m = 0; m < MN; ++m) {
        const unsigned row = (unsigned)__shfl(myIdx, m, 16);  // DPP row_share
        const unsigned off = row * (HID * 4u) + laneOff;      // 32-bit voffset

        const v4f v0 = *(const v4f*)(xb + off);               // saddr + voffset
        const v4f v1 = *(const v4f*)(xb + slab + off);
        const v4f v2 = *(const v4f*)(xb + 2 * slab + off);
        const v4f v3 = *(const v4f*)(xb + 3 * slab + off);

        a0 = vmax4(a0, v0);
        a1 = vmax4(a1, v1);
        a2 = vmax4(a2, v2);
        a3 = vmax4(a3, v3);
    }

    // Restore the zero-row contribution of any padded neighbor.
    v4f f; f.x = f.y = f.z = f.w = floorVal;
    a0 = vmax4(a0, f);
    a1 = vmax4(a1, f);
    a2 = vmax4(a2, f);
    a3 = vmax4(a3, f);

    __builtin_nontemporal_store(a0, (v4f*)(ob + outOff));
    __builtin_nontemporal_store(a1, (v4f*)(ob + slab + outOff));
    __builtin_nontemporal_store(a2, (v4f*)(ob + 2 * slab + outOff));
    __builtin_nontemporal_store(a3, (v4f*)(ob + 3 * slab + outOff));
}

extern "C" void kernel_launch(void* const* d_in, const int* in_sizes, int n_in,
                              void* d_out, int out_size, void* d_ws, size_t ws_size,
                              hipStream_t stream) {
    const float* x     = (const float*)d_in[0];
    const int*   index = (const int*)  d_in[1];
    float*       out   = (float*)      d_out;

    // in_sizes[1] == N * M_NEIGH
    const int N = in_sizes[1] / MN;

    dim3 block(256);
    dim3 grid((N + NODES_PER_BLOCK - 1) / NODES_PER_BLOCK);
    gather_max16_kernel<<<grid, block, 0, stream>>>(x, index, out, N);
}